// StructureAwareattenation_21311627722961
// MI455X (gfx1250) — compile-verified
//
#include <hip/hip_runtime.h>
#include <hip/hip_bf16.h>
#include <math.h>

typedef float v2f __attribute__((ext_vector_type(2)));
typedef float v4f __attribute__((ext_vector_type(4)));
typedef float v8f __attribute__((ext_vector_type(8)));

#define NB    2
#define NS    1024
#define NH    16
#define NHID  128
#define LUT_N 4096
#define TI    8        // rows of i per block in the main kernel

__device__ __forceinline__ float gelu_exact(float x) {
    // torch-default exact GELU: 0.5*x*(1+erf(x/sqrt(2)))
    return 0.5f * x * (1.0f + erff(x * 0.70710678118654752440f));
}

// -----------------------------------------------------------------------------
// Kernel 1: build LUT  g[t] = exp(-3*(w2 . gelu(d*w1 + b1) + b2)),  d = t/4095
// One wave per 16-entry tile. Layer-2 reduction (K=128) done with
// V_WMMA_F32_16X16X4_F32: A = gelu-activations (16 entries x 4 k), B = w2
// broadcast to all 16 columns, C accumulates f32.
//   A layout (f32 16x4):  lane L holds row M=L&15; VGPR0 = K = k0+2*(L>>4),
//                         VGPR1 = K+1.
//   B layout (f32 4x16):  lane L holds col N=L&15; VGPR0 = row K = k0+2*(L>>4),
//                         VGPR1 = row K+1.  (broadcast: value depends on K only)
//   D layout (f32 16x16): VGPR v, lane L -> M = v + 8*(L>>4), N = L&15.
// -----------------------------------------------------------------------------
__global__ void build_lut_wmma(const float* __restrict__ w1,
                               const float* __restrict__ b1,
                               const float* __restrict__ w2,
                               const float* __restrict__ b2,
                               float* __restrict__ g_lut) {
    const int lane = threadIdx.x;        // 0..31, full wave, EXEC all ones
    const int m    = lane & 15;
    const int half = lane >> 4;
    const int tile = blockIdx.x;         // 0..LUT_N/16-1
    const float d  = (float)(tile * 16 + m) * (1.0f / (float)(LUT_N - 1));

    v8f c = {};                          // f32 accumulator
    #pragma unroll 4
    for (int k0 = 0; k0 < NHID; k0 += 4) {
        const int ka = k0 + 2 * half;
        float pa = fmaf(d, w1[ka],     b1[ka]);
        float pb = fmaf(d, w1[ka + 1], b1[ka + 1]);
        v2f a;    a.x    = gelu_exact(pa);  a.y    = gelu_exact(pb);
        v2f bmat; bmat.x = w2[ka];          bmat.y = w2[ka + 1];
        c = __builtin_amdgcn_wmma_f32_16x16x4_f32(
                /*neg_a=*/false, a, /*neg_b=*/false, bmat,
                /*c_mod=*/(short)0, c, /*reuse_a=*/false, /*reuse_b=*/false);
    }

    // All 16 columns of D are identical (B columns identical); extract from N=0
    // lanes: lane 0 holds M=0..7 in c[0..7], lane 16 holds M=8..15.
    if ((lane & 15) == 0) {
        const float b2v  = b2[0];
        const int   base = tile * 16 + half * 8;
        #pragma unroll
        for (int v = 0; v < 8; ++v)
            g_lut[base + v] = expf(-3.0f * (c[v] + b2v));
    }
}

// -----------------------------------------------------------------------------
// Kernel 2: bandwidth-bound main pass.
// Block = 256 threads, covers (b, i0..i0+7, all 1024 j). Thread t handles
// j = 4t..4t+3 (128-bit stores, 512B contiguous per wave per store).
// dist -> clamp -> LDS-LUT lerp -> mask -> 16 head-broadcast NT stores.
// -----------------------------------------------------------------------------
__global__ void attn_lut_kernel(const float* __restrict__ coords,
                                const unsigned char* __restrict__ mask,
                                const float* __restrict__ g_lut,
                                float* __restrict__ out) {
    __shared__ float s_lut[LUT_N];                 // 16 KB
    __shared__ float s_ix[TI], s_iy[TI], s_iz[TI];
    __shared__ int   s_mi[TI];

    const int tid      = threadIdx.x;              // 0..255
    const int rowsPerB = NS / TI;                  // 128
    const int b        = blockIdx.x / rowsPerB;
    const int i0       = (blockIdx.x % rowsPerB) * TI;

    __builtin_prefetch(g_lut + tid * 16, 0, 0);    // global_prefetch_b8

    for (int t = tid; t < LUT_N; t += 256) s_lut[t] = g_lut[t];
    if (tid < TI) {
        const int i = i0 + tid;
        const float* cp = coords + ((size_t)b * NS + i) * 3;
        s_ix[tid] = cp[0]; s_iy[tid] = cp[1]; s_iz[tid] = cp[2];
        s_mi[tid] = (int)mask[b * NS + i];
    }
    __syncthreads();

    // 4 consecutive j per thread: 12 contiguous, 16B-aligned floats.
    const int j = tid * 4;
    const v4f* cj4 = (const v4f*)(coords + ((size_t)b * NS + j) * 3);
    const v4f c0 = cj4[0], c1 = cj4[1], c2 = cj4[2];
    float jx[4], jy[4], jz[4];
    jx[0] = c0.x; jy[0] = c0.y; jz[0] = c0.z;
    jx[1] = c0.w; jy[1] = c1.x; jz[1] = c1.y;
    jx[2] = c1.z; jy[2] = c1.w; jz[2] = c2.x;
    jx[3] = c2.y; jy[3] = c2.z; jz[3] = c2.w;
    int mj[4];
    #pragma unroll
    for (int q = 0; q < 4; ++q) mj[q] = (int)mask[b * NS + j + q];

    const size_t headStride = (size_t)NS * NS;     // 1M floats per head plane

    for (int ii = 0; ii < TI; ++ii) {
        const int   i  = i0 + ii;
        const float xi = s_ix[ii], yi = s_iy[ii], zi = s_iz[ii];
        const int   mi = s_mi[ii];
        v4f val;
        #pragma unroll
        for (int q = 0; q < 4; ++q) {
            const float dx = xi - jx[q], dy = yi - jy[q], dz = zi - jz[q];
            const float dist = sqrtf(fmaf(dx, dx, fmaf(dy, dy, fmaf(dz, dz, 1e-8f))));
            float u = dist * (1.0f / 20.0f);
            u = fminf(fmaxf(u, 0.0f), 1.0f);
            const float x = u * (float)(LUT_N - 1);
            int   t = (int)x;
            t = min(t, LUT_N - 2);
            const float f  = x - (float)t;
            const float g0 = s_lut[t];
            const float g1 = s_lut[t + 1];
            const float g  = fmaf(f, g1 - g0, g0);
            val[q] = (mi | mj[q]) ? -1e9f : g;
        }
        const size_t rowbase = (size_t)b * NH * headStride + (size_t)i * NS + j;
        #pragma unroll
        for (int h = 0; h < NH; ++h) {
            __builtin_nontemporal_store(
                val, (v4f*)(out + rowbase + (size_t)h * headStride));
        }
    }
}

// -----------------------------------------------------------------------------
extern "C" void kernel_launch(void* const* d_in, const int* in_sizes, int n_in,
                              void* d_out, int out_size, void* d_ws, size_t ws_size,
                              hipStream_t stream) {
    const float*         coords = (const float*)d_in[0];
    const unsigned char* pmask  = (const unsigned char*)d_in[1]; // bool array
    const float*         w1     = (const float*)d_in[2];         // [128,1]
    const float*         b1     = (const float*)d_in[3];         // [128]
    const float*         w2     = (const float*)d_in[4];         // [1,128]
    const float*         b2     = (const float*)d_in[5];         // [1]
    float* out   = (float*)d_out;
    float* g_lut = (float*)d_ws;                                 // 16 KB scratch

    build_lut_wmma<<<LUT_N / 16, 32, 0, stream>>>(w1, b1, w2, b2, g_lut);
    attn_lut_kernel<<<NB * (NS / TI), 256, 0, stream>>>(coords, pmask, g_lut, out);
}